// SPLINE_LAYER_33938831573701
// MI455X (gfx1250) — compile-verified
//
#include <hip/hip_runtime.h>
#include <math.h>

#define DIMX   64
#define NBINS  32
#define HIDN   100
#define CDIM   64
#define MT     16          // rows per workgroup
#define LEFTB  (-1.0f)
#define RIGHTB (1.0f)
#define SPANB  (2.0f)
#define EPSB   (0.01f)
#define EPSS   (0.01f)
#define CFAC   (1.0f - EPSB * (float)NBINS)   // 0.68

#define NWC    2048        // DIM*NB columns of w/h W2
#define NSC    2112        // DIM*(NB+1) columns of s W2
#define CHD    16          // dims per chunk (4 chunks)
#define WCHP   513         // padded LDS stride for w/h chunk (odd pad -> conflict-free spline reads)
#define SCHP   529         // padded LDS stride for s chunk
#define XP     68          // padded stride for x / logd tiles
#define CP     68          // padded stride for cond tile

typedef float v2f __attribute__((ext_vector_type(2)));
typedef float v8f __attribute__((ext_vector_type(8)));

__device__ __forceinline__ float softplus_eps(float v) {
    // EPS_SLOPE + softplus(v), numerically stable
    return EPSS + fmaxf(v, 0.0f) + log1pf(__expf(-fabsf(v)));
}

// One 16x16 output tile of stage-2 GEMM: C[16x16] += H[16x100] @ W2[100, cols@cm..cm+16]
// NCOL compile-time => all 50 global loads are base + immediate offset.
template<int NCOL, int LDP>
__device__ __forceinline__ void gemm_tile25(const v2f* __restrict__ areg,
                                            const float* __restrict__ bp,   // W2 + 2*hi*NCOL + cm
                                            const float* __restrict__ B2,
                                            float* __restrict__ dst,
                                            int cm, int clds, int nl, int hi)
{
    v8f acc0 = {}; v8f acc1 = {};
    #pragma unroll
    for (int ks = 0; ks < 25; ++ks) {
        v2f b;
        b.x = bp[ks * 4 * NCOL];
        b.y = bp[(ks * 4 + 1) * NCOL];
        if (ks & 1)
            acc1 = __builtin_amdgcn_wmma_f32_16x16x4_f32(false, areg[ks], false, b, (short)0, acc1, false, false);
        else
            acc0 = __builtin_amdgcn_wmma_f32_16x16x4_f32(false, areg[ks], false, b, (short)0, acc0, false, false);
    }
    v8f acc = acc0 + acc1;
    float bias = B2[cm];
    #pragma unroll
    for (int i = 0; i < 8; ++i)
        dst[(i + 8 * hi) * LDP + clds + nl] = acc[i] + bias;
}

__launch_bounds__(256)
__global__ void spline_fused(
    const float* __restrict__ x,    const float* __restrict__ cond,
    const int*   __restrict__ pmask,
    const float* __restrict__ w_W1, const float* __restrict__ w_b1,
    const float* __restrict__ w_W2, const float* __restrict__ w_b2,
    const float* __restrict__ h_W1, const float* __restrict__ h_b1,
    const float* __restrict__ h_W2, const float* __restrict__ h_b2,
    const float* __restrict__ s_W1, const float* __restrict__ s_b1,
    const float* __restrict__ s_W2, const float* __restrict__ s_b2,
    float* __restrict__ out, int mtot)
{
    __shared__ float lX[MT * XP];             // x tile
    __shared__ float lC[MT * CP];             // cond tile
    __shared__ float lHid[3 * MT * HIDN];     // silu hidden, 3 MLPs
    __shared__ float lW[MT * WCHP];           // w logits chunk (16 dims)
    __shared__ float lH[MT * WCHP];           // h logits chunk
    __shared__ float lS[MT * SCHP];           // s raw chunk
    __shared__ float lLd[MT * XP];            // per-(row,dim) log-deriv

    const int tid  = threadIdx.x;
    const int lane = tid & 31;
    const int wv   = tid >> 5;     // wave 0..7
    const int nl   = lane & 15;    // N index / M index within fragment
    const int hi   = lane >> 4;    // lane-half selector
    const int row0 = blockIdx.x * MT;

    // ---- Phase 0: stage x and cond tiles into LDS (coalesced) ----
    for (int i = tid; i < MT * DIMX; i += 256) {
        int r = i >> 6, c = i & 63;
        lX[r * XP + c] = x[(row0 + r) * DIMX + c];
        lC[r * CP + c] = cond[(row0 + r) * CDIM + c];
    }
    __syncthreads();

    // ---- Phase 1: hidden = silu(cond @ W1 + b1) for 3 MLPs ----
    {
        // A fragments (cond) are identical for all 21 tiles: load once (b64 LDS reads)
        v2f acond[16];
        #pragma unroll
        for (int ks = 0; ks < 16; ++ks)
            acond[ks] = *(const v2f*)&lC[nl * CP + ks * 4 + 2 * hi];

        for (int t = wv; t < 21; t += 8) {
            int mlp = t / 7;
            int n0  = (t % 7) * 16;
            int col = n0 + nl;
            int colc = (col < HIDN) ? col : (HIDN - 1);   // clamp: garbage cols never stored
            const float* W1 = (mlp == 0) ? w_W1 : (mlp == 1) ? h_W1 : s_W1;
            const float* B1 = (mlp == 0) ? w_b1 : (mlp == 1) ? h_b1 : s_b1;
            const float* bp = W1 + 2 * hi * HIDN + colc;  // base + immediate offsets below
            v8f acc0 = {}; v8f acc1 = {};
            #pragma unroll
            for (int ks = 0; ks < 16; ++ks) {
                v2f b;
                b.x = bp[ks * 4 * HIDN];
                b.y = bp[(ks * 4 + 1) * HIDN];
                if (ks & 1)
                    acc1 = __builtin_amdgcn_wmma_f32_16x16x4_f32(false, acond[ks], false, b, (short)0, acc1, false, false);
                else
                    acc0 = __builtin_amdgcn_wmma_f32_16x16x4_f32(false, acond[ks], false, b, (short)0, acc0, false, false);
            }
            v8f acc = acc0 + acc1;
            if (col < HIDN) {
                float bias = B1[col];
                #pragma unroll
                for (int i = 0; i < 8; ++i) {
                    int r = i + 8 * hi;
                    float v = acc[i] + bias;
                    lHid[(mlp * MT + r) * HIDN + col] = v / (1.0f + __expf(-v)); // silu
                }
            }
        }
    }
    __syncthreads();

    // ---- Phases 2+3: 4 chunks of 16 dims ----
    for (int ch = 0; ch < DIMX / CHD; ++ch) {
        int d0 = ch * CHD;

        // Phase 2: logits = hidden @ W2 + b2 for this chunk.
        // Three specialized segments so NCOL is compile-time and A fragments live in registers.
        {   // w segment: 32 tiles
            v2f areg[25];
            const float* Hb = &lHid[0];
            #pragma unroll
            for (int ks = 0; ks < 25; ++ks)
                areg[ks] = *(const v2f*)&Hb[nl * HIDN + ks * 4 + 2 * hi];
            for (int t = wv; t < 32; t += 8) {
                int cm = d0 * NBINS + t * 16 + nl;
                gemm_tile25<NWC, WCHP>(areg, w_W2 + 2 * hi * NWC + cm, w_b2, lW, cm, t * 16, nl, hi);
            }
        }
        {   // h segment: 32 tiles
            v2f areg[25];
            const float* Hb = &lHid[MT * HIDN];
            #pragma unroll
            for (int ks = 0; ks < 25; ++ks)
                areg[ks] = *(const v2f*)&Hb[nl * HIDN + ks * 4 + 2 * hi];
            for (int t = wv; t < 32; t += 8) {
                int cm = d0 * NBINS + t * 16 + nl;
                gemm_tile25<NWC, WCHP>(areg, h_W2 + 2 * hi * NWC + cm, h_b2, lH, cm, t * 16, nl, hi);
            }
        }
        {   // s segment: 33 tiles (column bases may be 16-unaligned: fine, per-lane addressing)
            v2f areg[25];
            const float* Hb = &lHid[2 * MT * HIDN];
            #pragma unroll
            for (int ks = 0; ks < 25; ++ks)
                areg[ks] = *(const v2f*)&Hb[nl * HIDN + ks * 4 + 2 * hi];
            for (int t = wv; t < 33; t += 8) {
                int cm = d0 * (NBINS + 1) + t * 16 + nl;
                gemm_tile25<NSC, SCHP>(areg, s_W2 + 2 * hi * NSC + cm, s_b2, lS, cm, t * 16, nl, hi);
            }
        }
        __syncthreads();

        // Phase 3: RQ spline; one thread per (row, dim) pair — 256 pairs
        {
            int r   = tid & 15;
            int dl  = tid >> 4;
            int dim = d0 + dl;
            float xv = lX[r * XP + dim];
            float* pW = &lW[r * WCHP + dl * NBINS];
            float* pH = &lH[r * WCHP + dl * NBINS];
            float* pS = &lS[r * SCHP + dl * (NBINS + 1)];

            // softmax numerators cached back into LDS
            float mw = pW[0];
            for (int j = 1; j < NBINS; ++j) mw = fmaxf(mw, pW[j]);
            float sumw = 0.0f;
            for (int j = 0; j < NBINS; ++j) { float e = __expf(pW[j] - mw); pW[j] = e; sumw += e; }
            float mh = pH[0];
            for (int j = 1; j < NBINS; ++j) mh = fmaxf(mh, pH[j]);
            float sumh = 0.0f;
            for (int j = 0; j < NBINS; ++j) { float e = __expf(pH[j] - mh); pH[j] = e; sumh += e; }
            float fw = CFAC / sumw, fh = CFAC / sumh;

            // cumsum + bin search on widths (endpoints forced to LEFT/RIGHT like reference)
            float cw = LEFTB, cwk = LEFTB, wk = SPANB; int idx = NBINS - 1; bool found = false;
            for (int k = 0; k < NBINS; ++k) {
                float nxt = (k == NBINS - 1) ? RIGHTB : cw + SPANB * (EPSB + pW[k] * fw);
                if (!found && (xv < nxt || k == NBINS - 1)) { idx = k; cwk = cw; wk = nxt - cw; found = true; }
                cw = nxt;
            }
            // cumsum on heights, gather at idx
            float chc = LEFTB, chk = LEFTB, hk = SPANB;
            for (int k = 0; k < NBINS; ++k) {
                float nxt = (k == NBINS - 1) ? RIGHTB : chc + SPANB * (EPSB + pH[k] * fh);
                if (k == idx) { chk = chc; hk = nxt - chc; }
                chc = nxt;
            }
            // derivatives with periodic boundary tie
            float dk  = softplus_eps(pS[idx]);
            bool per  = (pmask[dim] != 0);
            float sv1 = (idx == NBINS - 1 && per) ? pS[0] : pS[idx + 1];
            float dk1 = softplus_eps(sv1);

            float delta = hk / wk;
            float th    = (xv - cwk) / wk;
            float omt   = 1.0f - th;
            float t1m   = th * omt;
            float den   = delta + (dk1 + dk - 2.0f * delta) * t1m;
            float yv    = chk + hk * (delta * th * th + dk * t1m) / den;
            float dv    = (delta * delta) * (dk1 * th * th + 2.0f * delta * t1m + dk * omt * omt) / (den * den);

            out[(row0 + r) * DIMX + dim] = yv;
            lLd[r * XP + dim] = __logf(dv);
        }
        __syncthreads();
    }

    // ---- ladJ reduction: sum log(deriv) over 64 dims per row ----
    if (tid < MT) {
        float s = 0.0f;
        for (int j = 0; j < DIMX; ++j) s += lLd[tid * XP + j];
        out[(size_t)mtot * DIMX + row0 + tid] = s;
    }
}

extern "C" void kernel_launch(void* const* d_in, const int* in_sizes, int n_in,
                              void* d_out, int out_size, void* d_ws, size_t ws_size,
                              hipStream_t stream) {
    (void)n_in; (void)out_size; (void)d_ws; (void)ws_size;
    const float* x    = (const float*)d_in[0];
    const float* cond = (const float*)d_in[1];
    const int*   pm   = (const int*)d_in[2];
    const float* wW1  = (const float*)d_in[3];  const float* wb1 = (const float*)d_in[4];
    const float* wW2  = (const float*)d_in[5];  const float* wb2 = (const float*)d_in[6];
    const float* hW1  = (const float*)d_in[7];  const float* hb1 = (const float*)d_in[8];
    const float* hW2  = (const float*)d_in[9];  const float* hb2 = (const float*)d_in[10];
    const float* sW1  = (const float*)d_in[11]; const float* sb1 = (const float*)d_in[12];
    const float* sW2  = (const float*)d_in[13]; const float* sb2 = (const float*)d_in[14];
    float* out = (float*)d_out;
    int m = in_sizes[0] / DIMX;          // 16384
    dim3 grid((m + MT - 1) / MT);        // 1024 workgroups of 16 rows
    spline_fused<<<grid, 256, 0, stream>>>(x, cond, pm,
                                           wW1, wb1, wW2, wb2,
                                           hW1, hb1, hW2, hb2,
                                           sW1, sb1, sW2, sb2,
                                           out, m);
}